// BLAMem_80169859547641
// MI455X (gfx1250) — compile-verified
//
#include <hip/hip_runtime.h>

// ---------------------------------------------------------------------------
// Log-signature memory forward pass for MI455X (gfx1250, wave32).
// Tensor-algebra ops (rank-1 outer-product accumulations) -> VALU + LDS.
// Element staging (19KB/element) -> GLOBAL_LOAD_ASYNC_TO_LDS_B128 (ASYNCcnt).
// Dense GEMM pooled(16x4680) @ W1(4680x256) -> v_wmma_f32_16x16x4_f32.
// ---------------------------------------------------------------------------

#define NTH 256

constexpr int BATCH = 8;
constexpr int TLEN  = 2048;
constexpr int CIN   = 7;
constexpr int NCH   = 128;   // TLEN / 16 chunks
constexpr int OFF1 = 0, OFF2 = 8, OFF3 = 72, OFF4 = 584;
constexpr int SIGD = 4680;   // 8 + 64 + 512 + 4096  (= 1170 * 4, 16B-aligned)
constexpr int SIGQ = SIGD / 4;
constexpr int HDIM = 256;

typedef __attribute__((ext_vector_type(2))) float v2f;
typedef __attribute__((ext_vector_type(8))) float v8f;
typedef __attribute__((ext_vector_type(4))) int   v4i;

#if defined(__has_builtin)
#if __has_builtin(__builtin_amdgcn_global_load_async_to_lds_b128)
#define USE_ASYNC_LDS 1
#else
#define USE_ASYNC_LDS 0
#endif
#else
#define USE_ASYNC_LDS 0
#endif

// ---- memory -> LDS element staging ----------------------------------------

__device__ inline void copy_elem_to_lds(const float* __restrict__ src,
                                        float* __restrict__ dst, int tid) {
#if USE_ASYNC_LDS
  for (int q = tid; q < SIGQ; q += NTH) {
    __builtin_amdgcn_global_load_async_to_lds_b128(
        (__attribute__((address_space(1))) v4i*)(v4i*)(src + 4 * q),
        (__attribute__((address_space(3))) v4i*)(v4i*)(dst + 4 * q),
        /*offset=*/0, /*cpol=*/0);
  }
#else
  for (int q = tid; q < SIGQ; q += NTH)
    ((float4*)dst)[q] = ((const float4*)src)[q];
#endif
}

__device__ inline void stage_join() {
#if USE_ASYNC_LDS
#if __has_builtin(__builtin_amdgcn_s_wait_asynccnt)
  __builtin_amdgcn_s_wait_asynccnt(0);
#else
  asm volatile("s_wait_asynccnt 0x0" ::: "memory");
#endif
#endif
  __syncthreads();
}

// ---- device helpers on LDS-resident 4680-float elements -------------------

// P <- ta_mulz(S, P) in place. Levels written high->low; level k reads only
// levels < k of P, so in-place is safe with a barrier between levels.
__device__ inline void mulz_inplace(const float* __restrict__ S,
                                    float* __restrict__ P, int tid) {
  for (int idx = tid; idx < 4096; idx += NTH) {
    P[OFF4 + idx] = S[OFF1 + (idx >> 9)] * P[OFF3 + (idx & 511)]
                  + S[OFF2 + (idx >> 6)] * P[OFF2 + (idx & 63)]
                  + S[OFF3 + (idx >> 3)] * P[OFF1 + (idx & 7)];
  }
  __syncthreads();
  for (int idx = tid; idx < 512; idx += NTH) {
    P[OFF3 + idx] = S[OFF1 + (idx >> 6)] * P[OFF2 + (idx & 63)]
                  + S[OFF2 + (idx >> 3)] * P[OFF1 + (idx & 7)];
  }
  __syncthreads();
  if (tid < 64)
    P[OFF2 + tid] = S[OFF1 + (tid >> 3)] * P[OFF1 + (tid & 7)];
  __syncthreads();
  if (tid < 8) P[OFF1 + tid] = 0.0f;
  __syncthreads();
}

// O <- A + c2*A^2 + c3*A^3 + c4*A^4  (mulz powers); serves as ta_exp / ta_log.
__device__ inline void powser_lds(const float* __restrict__ A,
                                  float* __restrict__ P, float* __restrict__ O,
                                  float c2, float c3, float c4, int tid) {
  for (int i = tid; i < SIGD; i += NTH) { float v = A[i]; P[i] = v; O[i] = v; }
  __syncthreads();
  const float cs[3] = {c2, c3, c4};
  for (int m = 0; m < 3; ++m) {
    mulz_inplace(A, P, tid);
    const float c = cs[m];
    for (int i = tid; i < SIGD; i += NTH) O[i] += c * P[i];
    __syncthreads();
  }
}

// R <- A (x) B, group-like product (implicit level-0 == 1 on both sides).
__device__ inline void ta_mul_lds(const float* __restrict__ A,
                                  const float* __restrict__ Bv,
                                  float* __restrict__ R, int tid) {
  for (int idx = tid; idx < 4096; idx += NTH) {
    R[OFF4 + idx] = A[OFF4 + idx] + Bv[OFF4 + idx]
                  + A[OFF1 + (idx >> 9)] * Bv[OFF3 + (idx & 511)]
                  + A[OFF2 + (idx >> 6)] * Bv[OFF2 + (idx & 63)]
                  + A[OFF3 + (idx >> 3)] * Bv[OFF1 + (idx & 7)];
  }
  for (int idx = tid; idx < 512; idx += NTH) {
    R[OFF3 + idx] = A[OFF3 + idx] + Bv[OFF3 + idx]
                  + A[OFF1 + (idx >> 6)] * Bv[OFF2 + (idx & 63)]
                  + A[OFF2 + (idx >> 3)] * Bv[OFF1 + (idx & 7)];
  }
  if (tid < 64)
    R[OFF2 + tid] = A[OFF2 + tid] + Bv[OFF2 + tid]
                  + A[OFF1 + (tid >> 3)] * Bv[OFF1 + (tid & 7)];
  if (tid < 8)
    R[OFF1 + tid] = A[OFF1 + tid] + Bv[OFF1 + tid];
  __syncthreads();
}

// ---------------- kernel 1: per-chunk signature via Chen -------------------

__global__ __launch_bounds__(NTH) void k_chunk_sig(const float* __restrict__ x,
                                                   float* __restrict__ sig) {
  __shared__ __attribute__((aligned(16))) float S[SIGD];
  __shared__ float dinc[16][8];
  const int tid = threadIdx.x;
  const int p = blockIdx.x;
  const int b = p >> 7, n = p & (NCH - 1);

  // increments of [x, t] with basepoint-0 prepend; time inc = 1/2047 (0 at t=0)
  if (tid < 128) {
    const int l = tid >> 3, c = tid & 7;
    const int t = n * 16 + l;
    float v;
    if (c == 7) {
      v = (t == 0) ? 0.0f : (1.0f / 2047.0f);
    } else {
      const float cur  = x[(b * TLEN + t) * CIN + c];
      const float prev = (t > 0) ? x[(b * TLEN + t - 1) * CIN + c] : 0.0f;
      v = cur - prev;
    }
    dinc[l][c] = v;
  }
  for (int i = tid; i < SIGD; i += NTH) S[i] = 0.0f;
  __syncthreads();

  // S <- S (x) exp_l1(d_l), l = 0..15; in place, levels high->low.
  for (int l = 0; l < 16; ++l) {
    float rd[8];
#pragma unroll
    for (int c = 0; c < 8; ++c) rd[c] = dinc[l][c];

    for (int idx = tid; idx < 4096; idx += NTH) {
      const int i = idx >> 9, j = (idx >> 6) & 7, k = (idx >> 3) & 7, m = idx & 7;
      const float dj = rd[j], dk = rd[k], dm = rd[m], di = rd[i];
      S[OFF4 + idx] = S[OFF4 + idx]
                    + S[OFF3 + (idx >> 3)] * dm
                    + S[OFF2 + (idx >> 6)] * (dk * dm * 0.5f)
                    + S[OFF1 + i] * (dj * dk * dm * (1.0f / 6.0f))
                    + di * dj * dk * dm * (1.0f / 24.0f);
    }
    __syncthreads();
    for (int idx = tid; idx < 512; idx += NTH) {
      const int i = idx >> 6, j = (idx >> 3) & 7, k = idx & 7;
      S[OFF3 + idx] = S[OFF3 + idx]
                    + S[OFF2 + (idx >> 3)] * rd[k]
                    + S[OFF1 + i] * (rd[j] * rd[k] * 0.5f)
                    + rd[i] * rd[j] * rd[k] * (1.0f / 6.0f);
    }
    __syncthreads();
    if (tid < 64) {
      const int i = tid >> 3, j = tid & 7;
      S[OFF2 + tid] += S[OFF1 + i] * rd[j] + rd[i] * rd[j] * 0.5f;
    }
    __syncthreads();
    if (tid < 8) S[OFF1 + tid] += rd[tid];
    __syncthreads();
  }

  float4* out = (float4*)(sig + (size_t)p * SIGD);
  for (int q = tid; q < SIGQ; q += NTH) out[q] = ((const float4*)S)[q];
}

// ---------------- kernel 2: elementwise power series (exp / log) -----------

__global__ __launch_bounds__(NTH) void k_powser(const float* __restrict__ in,
                                                float* __restrict__ out,
                                                float c2, float c3, float c4) {
  __shared__ __attribute__((aligned(16))) float A[SIGD];
  __shared__ __attribute__((aligned(16))) float P[SIGD];
  __shared__ __attribute__((aligned(16))) float O[SIGD];
  const int tid = threadIdx.x;
  const size_t base = (size_t)blockIdx.x * SIGD;
  copy_elem_to_lds(in + base, A, tid);
  stage_join();
  powser_lds(A, P, O, c2, c3, c4, tid);
  float4* dst = (float4*)(out + base);
  for (int q = tid; q < SIGQ; q += NTH) dst[q] = ((const float4*)O)[q];
}

// ---------------- kernel 3: BCH merge body: log(E[n-d] (x) E[n]) -----------

__global__ __launch_bounds__(NTH) void k_mullog(const float* __restrict__ E,
                                                const float* __restrict__ pref,
                                                float* __restrict__ out, int d) {
  __shared__ __attribute__((aligned(16))) float B0[SIGD];
  __shared__ __attribute__((aligned(16))) float B1[SIGD];
  __shared__ __attribute__((aligned(16))) float B2[SIGD];
  const int tid = threadIdx.x;
  const int p = blockIdx.x;
  const int n = p & (NCH - 1);
  float* dst = out + (size_t)p * SIGD;
  if (n < d) {  // bch(0, b) == b exactly
    const float4* src4 = (const float4*)(pref + (size_t)p * SIGD);
    float4* dst4 = (float4*)dst;
    for (int q = tid; q < SIGQ; q += NTH) dst4[q] = src4[q];
    return;
  }
  copy_elem_to_lds(E + (size_t)(p - d) * SIGD, B0, tid);
  copy_elem_to_lds(E + (size_t)p * SIGD, B1, tid);
  stage_join();
  ta_mul_lds(B0, B1, B2, tid);                              // B2 = exp(a)(x)exp(b)
  powser_lds(B2, B0, B1, -0.5f, 1.0f / 3.0f, -0.25f, tid);  // B1 = log(B2)
  float4* dst4 = (float4*)dst;
  for (int q = tid; q < SIGQ; q += NTH) dst4[q] = ((const float4*)B1)[q];
}

// ---------------- kernel 4: mean-pool over chunks, zero-pad M to 16 --------

__global__ void k_pool(const float* __restrict__ logpref,
                       float* __restrict__ pooled) {
  const int idx = blockIdx.x * blockDim.x + threadIdx.x;
  if (idx >= 16 * SIGD) return;
  const int b = idx / SIGD, f = idx - b * SIGD;
  float s = 0.0f;
  if (b < BATCH) {
    for (int n = 0; n < NCH; ++n)
      s += logpref[((size_t)(b * NCH + n)) * SIGD + f];
    s *= (1.0f / (float)NCH);
  }
  pooled[idx] = s;
}

// ---------------- kernel 5: MLP head with v_wmma_f32_16x16x4_f32 -----------

__global__ __launch_bounds__(512) void k_mlp(const float* __restrict__ pooled,
                                             const float* __restrict__ W1,
                                             const float* __restrict__ b1,
                                             const float* __restrict__ W2,
                                             const float* __restrict__ b2,
                                             float* __restrict__ outp) {
  __shared__ float hbuf[BATCH * HDIM];
  const int tid  = threadIdx.x;
  const int wave = tid >> 5;        // 16 waves, one 16x16 tile each
  const int lane = tid & 31;        // wave32
  const int row  = lane & 15;       // A row / D column-within-tile
  const int half = lane >> 4;       // 0: K=0,1   1: K=2,3  (ISA 16x4 f32 A layout)
  const int colbase = wave * 16;

  v8f acc = (v8f)0.0f;
  for (int k = 0; k < SIGD; k += 4) {
    const int k0 = k + 2 * half;
    v2f a, bf;
    a.x  = pooled[row * SIGD + k0];
    a.y  = pooled[row * SIGD + k0 + 1];
    bf.x = W1[(size_t)k0 * HDIM + colbase + row];
    bf.y = W1[(size_t)(k0 + 1) * HDIM + colbase + row];
    acc = __builtin_amdgcn_wmma_f32_16x16x4_f32(
        /*neg_a=*/false, a, /*neg_b=*/false, bf,
        /*c_mod=*/(short)0, acc, /*reuse_a=*/false, /*reuse_b=*/false);
  }
  // D layout: VGPR r -> rows r (lanes 0-15) and r+8 (lanes 16-31); col = lane&15.
  if (half == 0) {
#pragma unroll
    for (int r = 0; r < 8; ++r) {
      const int ncol = colbase + row;
      const float h = acc[r] + b1[ncol];
      hbuf[r * HDIM + ncol] = h > 0.0f ? h : 0.0f;
    }
  }
  __syncthreads();
  if (tid < BATCH) {
    float s = b2[0];
    for (int j = 0; j < HDIM; ++j) s += hbuf[tid * HDIM + j] * W2[j];
    outp[tid] = s;
  }
}

// ---------------------------------------------------------------------------

extern "C" void kernel_launch(void* const* d_in, const int* in_sizes, int n_in,
                              void* d_out, int out_size, void* d_ws, size_t ws_size,
                              hipStream_t stream) {
  (void)in_sizes; (void)n_in; (void)out_size; (void)ws_size;
  const float* x  = (const float*)d_in[0];
  const float* W1 = (const float*)d_in[1];
  const float* b1 = (const float*)d_in[2];
  const float* W2 = (const float*)d_in[3];
  const float* b2 = (const float*)d_in[4];
  float* outp = (float*)d_out;
  float* ws   = (float*)d_ws;

  const size_t PR = (size_t)BATCH * NCH * SIGD;  // 4,792,320 floats
  float* bufA   = ws;                 // ping
  float* bufB   = ws + PR;            // pong
  float* bufE   = ws + 2 * PR;        // exp-domain scratch
  float* pooled = ws + 3 * PR;        // 16 x 4680 (zero-padded rows 8..15)

  const dim3 gElem(BATCH * NCH), bElem(NTH);

  // 1) chunk signatures (group domain)
  k_chunk_sig<<<gElem, bElem, 0, stream>>>(x, bufA);
  // 2) per-chunk truncated log:  bufB = ta_log(bufA)
  k_powser<<<gElem, bElem, 0, stream>>>(bufA, bufB, -0.5f, 1.0f / 3.0f, -0.25f);

  // 3) Hillis-Steele inclusive scan with BCH merge (7 rounds)
  float* cur = bufB;
  float* oth = bufA;
  for (int d = 1; d < NCH; d <<= 1) {
    // exp of every element (log -> group domain)
    k_powser<<<gElem, bElem, 0, stream>>>(cur, bufE, 0.5f, 1.0f / 6.0f, 1.0f / 24.0f);
    // merge: oth[n] = (n<d) ? cur[n] : log(E[n-d] (x) E[n])
    k_mullog<<<gElem, bElem, 0, stream>>>(bufE, cur, oth, d);
    float* t = cur; cur = oth; oth = t;
  }

  // 4) mean-pool over chunks into 16-row zero-padded matrix
  k_pool<<<dim3((16 * SIGD + NTH - 1) / NTH), bElem, 0, stream>>>(cur, pooled);
  // 5) MLP head (WMMA GEMM + GEMV)
  k_mlp<<<dim3(1), dim3(512), 0, stream>>>(pooled, W1, b1, W2, b2, outp);
}